// MambaBackbone_25692494364975
// MI455X (gfx1250) — compile-verified
//
#include <hip/hip_runtime.h>
#include <math.h>

// Problem constants (from reference)
constexpr int kNL = 12;
constexpr int kD  = 768;
constexpr int kL  = 1024;
constexpr int kB  = 2;
constexpr int kDI = 1536;     // 2*D
constexpr int kN  = 16;
constexpr int kK  = 4;        // conv taps
constexpr int kR  = 48;       // ceil(D/16)
constexpr int kPC = kR + 2 * kN;   // proj cols = 80
constexpr int kM  = kB * kL;       // 2048 rows of activations

typedef float v2f __attribute__((ext_vector_type(2)));
typedef float v8f __attribute__((ext_vector_type(8)));

__device__ __forceinline__ v8f wmma_f32(v2f a, v2f b, v8f c)
{
    return __builtin_amdgcn_wmma_f32_16x16x4_f32(
        false, a, false, b, (short)0, c, false, false);
}

// ---------------------------------------------------------------------------
// fp32 WMMA GEMM, all dims compile-time.  Wave tile 32x32: 4 accumulators,
// 2 A frags x 2 B frags -> 4 V_WMMA_F32_16X16X4_F32 per k-step.  8 waves
// arranged 2(M) x 4(N) -> 64 x 128 macro tile.  Requires M%64==0, N%128==0,
// KC%4==0 (true for all call sites) -> no bounds checks in the hot kernel.
//
// Fragment layouts per CDNA5 ISA 7.12.2:
//   A 16x4 : lane = 16*half + m ; VGPR0 = A[m][k0+2*half], VGPR1 = +1
//   B 4x16 : lane = 16*half + n ; VGPR0 = B[k0+2*half][n], VGPR1 = +1 row
//   C 16x16: VGPR i = C[m = i + 8*half][n = lane&15]
// ---------------------------------------------------------------------------
template <int KC, int LDA, int LDB, int LDC>
__global__ void gemm32x32(const float* __restrict__ A,
                          const float* __restrict__ Bw,
                          float* __restrict__ C)
{
    const int lane = threadIdx.x & 31;
    const int wave = threadIdx.x >> 5;              // 0..7
    const int tileM = blockIdx.x * 64  + (wave & 1) * 32;
    const int tileN = blockIdx.y * 128 + (wave >> 1) * 32;
    const int half = lane >> 4;                     // 0 or 1
    const int r    = lane & 15;

    v8f acc00 = {}; v8f acc01 = {}; v8f acc10 = {}; v8f acc11 = {};

    const float* a0 = A + (size_t)(tileM + r) * LDA + 2 * half;   // rows 0..15
    const float* a1 = a0 + (size_t)16 * LDA;                      // rows 16..31
    const float* b0 = Bw + (size_t)(2 * half) * LDB + tileN + r;  // cols 0..15 (+16)

    for (int k0 = 0; k0 < KC; k0 += 4) {
        v2f av0, av1, bv0, bv1;
        av0.x = a0[0];       av0.y = a0[1];
        av1.x = a1[0];       av1.y = a1[1];
        bv0.x = b0[0];       bv0.y = b0[LDB];
        bv1.x = b0[16];      bv1.y = b0[LDB + 16];
        __builtin_prefetch(b0 + 16 * LDB, 0, 1);    // global_prefetch_b8
        acc00 = wmma_f32(av0, bv0, acc00);
        acc01 = wmma_f32(av0, bv1, acc01);
        acc10 = wmma_f32(av1, bv0, acc10);
        acc11 = wmma_f32(av1, bv1, acc11);
        a0 += 4; a1 += 4; b0 += (size_t)4 * LDB;
    }

    float* c00 = C + (size_t)(tileM + 8 * half) * LDC + tileN + r;
    float* c10 = c00 + (size_t)16 * LDC;
#pragma unroll
    for (int i = 0; i < 8; ++i) {
        c00[(size_t)i * LDC]      = acc00[i];
        c00[(size_t)i * LDC + 16] = acc01[i];
        c10[(size_t)i * LDC]      = acc10[i];
        c10[(size_t)i * LDC + 16] = acc11[i];
    }
}

// Skinny-N variant: wave tile 32x16, whole-wave uniform guard (used for the
// (2048 x 1536) @ (1536 x 80) projection GEMM).
template <int Nv, int KC, int LDA, int LDB, int LDC>
__global__ void gemm32x16(const float* __restrict__ A,
                          const float* __restrict__ Bw,
                          float* __restrict__ C)
{
    const int lane = threadIdx.x & 31;
    const int wave = threadIdx.x >> 5;              // 0..7
    const int tileM = blockIdx.x * 64 + (wave & 1) * 32;
    const int tileN = blockIdx.y * 64 + (wave >> 1) * 16;
    if (tileN >= Nv) return;                        // whole-wave uniform

    const int half = lane >> 4;
    const int r    = lane & 15;

    v8f acc0 = {}; v8f acc1 = {};

    const float* a0 = A + (size_t)(tileM + r) * LDA + 2 * half;
    const float* a1 = a0 + (size_t)16 * LDA;
    const float* b0 = Bw + (size_t)(2 * half) * LDB + tileN + r;

    for (int k0 = 0; k0 < KC; k0 += 4) {
        v2f av0, av1, bv;
        av0.x = a0[0];  av0.y = a0[1];
        av1.x = a1[0];  av1.y = a1[1];
        bv.x  = b0[0];  bv.y  = b0[LDB];
        acc0 = wmma_f32(av0, bv, acc0);
        acc1 = wmma_f32(av1, bv, acc1);
        a0 += 4; a1 += 4; b0 += (size_t)4 * LDB;
    }

    float* c0 = C + (size_t)(tileM + 8 * half) * LDC + tileN + r;
    float* c1 = c0 + (size_t)16 * LDC;
#pragma unroll
    for (int i = 0; i < 8; ++i) {
        c0[(size_t)i * LDC] = acc0[i];
        c1[(size_t)i * LDC] = acc1[i];
    }
}

// ---------------------------------------------------------------------------
// residual += hidden ; out = layernorm(residual, w, b).  One block per row.
// ---------------------------------------------------------------------------
__global__ void add_layernorm(const float* __restrict__ h,
                              float* __restrict__ resid,
                              const float* __restrict__ w,
                              const float* __restrict__ b,
                              float* __restrict__ out)
{
    const int row = blockIdx.x;
    const float* hr = h     + (size_t)row * kD;
    float*       rr = resid + (size_t)row * kD;
    float*       oo = out   + (size_t)row * kD;

    __shared__ float red0[8];
    __shared__ float red1[8];

    float sum = 0.f, sq = 0.f;
    for (int i = threadIdx.x; i < kD; i += blockDim.x) {
        float v = hr[i] + rr[i];
        rr[i] = v;
        sum += v;
        sq  += v * v;
    }
    for (int off = 16; off > 0; off >>= 1) {        // wave32 reduce
        sum += __shfl_down(sum, off, 32);
        sq  += __shfl_down(sq,  off, 32);
    }
    const int wave = threadIdx.x >> 5;
    const int lane = threadIdx.x & 31;
    if (lane == 0) { red0[wave] = sum; red1[wave] = sq; }
    __syncthreads();
    if (threadIdx.x == 0) {
        float s = 0.f, q = 0.f;
        const int nw = blockDim.x >> 5;
        for (int i = 0; i < nw; ++i) { s += red0[i]; q += red1[i]; }
        red0[0] = s; red1[0] = q;
    }
    __syncthreads();
    const float mean = red0[0] / (float)kD;
    const float var  = red1[0] / (float)kD - mean * mean;
    const float rstd = rsqrtf(var + 1e-5f);
    for (int i = threadIdx.x; i < kD; i += blockDim.x)
        oo[i] = (rr[i] - mean) * rstd * w[i] + b[i];
}

// ---------------------------------------------------------------------------
// Depthwise causal conv (K=4) over L on xc = xz[..., :DI], + bias, SiLU.
// ---------------------------------------------------------------------------
__global__ void conv_silu(const float* __restrict__ xz,
                          const float* __restrict__ cw,   // (DI,4)
                          const float* __restrict__ cb,   // (DI)
                          float* __restrict__ xc)         // (B,L,DI)
{
    const size_t idx = (size_t)blockIdx.x * blockDim.x + threadIdx.x;
    const int d  = (int)(idx % kDI);
    const int l  = (int)((idx / kDI) % kL);
    const int bb = (int)(idx / ((size_t)kDI * kL));

    float acc = cb[d];
#pragma unroll
    for (int k = 0; k < kK; ++k) {
        const int ls = l + k - (kK - 1);
        if (ls >= 0)
            acc += xz[((size_t)(bb * kL + ls)) * (2 * kDI) + d] * cw[d * kK + k];
    }
    xc[idx] = acc / (1.f + expf(-acc));   // SiLU
}

// delta = softplus(delta_raw + b_dt[d])  (in place)
__global__ void softplus_bias(float* __restrict__ delta,
                              const float* __restrict__ bdt)
{
    const size_t idx = (size_t)blockIdx.x * blockDim.x + threadIdx.x;
    const int d = (int)(idx % kDI);
    const float v = delta[idx] + bdt[d];
    delta[idx] = fmaxf(v, 0.f) + log1pf(expf(-fabsf(v)));
}

// yb *= silu(z) where z = xz[..., DI:]
__global__ void gate_silu(float* __restrict__ yb,
                          const float* __restrict__ xz)
{
    const size_t idx = (size_t)blockIdx.x * blockDim.x + threadIdx.x;
    const int d = (int)(idx % kDI);
    const size_t row = idx / kDI;
    const float z = xz[row * (2 * kDI) + kDI + d];
    yb[idx] *= z / (1.f + expf(-z));
}

// ---------------------------------------------------------------------------
// Selective scan: one thread per (b,d) channel, recurrence over l with N=16
// states in registers.  Bm/Cm are 32 contiguous floats per step (proj cols
// 48..79, 16B aligned) -> 8 float4 broadcast loads per step; no barriers.
// Grid: (DI/256, B), block 256.
// ---------------------------------------------------------------------------
__global__ void selective_scan(const float* __restrict__ delta, // (B,L,DI)
                               const float* __restrict__ xc,    // (B,L,DI)
                               const float* __restrict__ proj,  // (B,L,80)
                               const float* __restrict__ Alog,  // (DI,N)
                               const float* __restrict__ Dsk,   // (DI)
                               float* __restrict__ y)           // (B,L,DI)
{
    const int bb = blockIdx.y;
    const int d  = blockIdx.x * blockDim.x + threadIdx.x;

    float aneg[kN], h[kN];
#pragma unroll
    for (int n = 0; n < kN; ++n) {
        aneg[n] = -expf(Alog[(size_t)d * kN + n]);
        h[n] = 0.f;
    }
    const float dsk = Dsk[d];

    const float* dp = delta + (size_t)bb * kL * kDI + d;
    const float* xp = xc    + (size_t)bb * kL * kDI + d;
    const float* pp = proj  + (size_t)bb * kL * kPC + kR;
    float*       yp = y     + (size_t)bb * kL * kDI + d;

    for (int l = 0; l < kL; ++l) {
        const float4* p4 = (const float4*)pp;
        float4 bq[4], cq[4];
#pragma unroll
        for (int j = 0; j < 4; ++j) { bq[j] = p4[j]; cq[j] = p4[4 + j]; }
        const float* Bs = (const float*)bq;
        const float* Cs = (const float*)cq;

        const float dl = *dp;
        const float xl = *xp;
        const float du = dl * xl;
        float acc = 0.f;
#pragma unroll
        for (int n = 0; n < kN; ++n) {
            h[n] = expf(dl * aneg[n]) * h[n] + du * Bs[n];
            acc += h[n] * Cs[n];
        }
        *yp = acc + xl * dsk;

        dp += kDI; xp += kDI; yp += kDI; pp += kPC;
    }
}

__global__ void zero_kernel(float* __restrict__ p, size_t n)
{
    const size_t idx = (size_t)blockIdx.x * blockDim.x + threadIdx.x;
    if (idx < n) p[idx] = 0.f;
}

// ---------------------------------------------------------------------------
// Launch
// ---------------------------------------------------------------------------
extern "C" void kernel_launch(void* const* d_in, const int* in_sizes, int n_in,
                              void* d_out, int out_size, void* d_ws, size_t ws_size,
                              hipStream_t stream)
{
    (void)in_sizes; (void)n_in; (void)out_size; (void)ws_size;

    const float* x      = (const float*)d_in[0];   // (B,L,D)
    const float* ln_w   = (const float*)d_in[1];   // (NL,D)
    const float* ln_b   = (const float*)d_in[2];   // (NL,D)
    const float* W_in   = (const float*)d_in[3];   // (NL,D,2DI)
    const float* conv_w = (const float*)d_in[4];   // (NL,DI,K)
    const float* conv_b = (const float*)d_in[5];   // (NL,DI)
    const float* W_x    = (const float*)d_in[6];   // (NL,DI,80)
    const float* W_dt   = (const float*)d_in[7];   // (NL,R,DI)
    const float* b_dt   = (const float*)d_in[8];   // (NL,DI)
    const float* A_log  = (const float*)d_in[9];   // (NL,DI,N)
    const float* D_skip = (const float*)d_in[10];  // (NL,DI)
    const float* W_out  = (const float*)d_in[11];  // (NL,DI,D)
    const float* lnf_w  = (const float*)d_in[12];  // (D)
    const float* lnf_b  = (const float*)d_in[13];  // (D)
    float* out = (float*)d_out;

    // Workspace layout (floats)
    const size_t nBLD  = (size_t)kB * kL * kD;    // 1,572,864
    const size_t nBLDI = (size_t)kB * kL * kDI;   // 3,145,728
    float* ws    = (float*)d_ws;
    float* resid = ws;
    float* xln   = resid + nBLD;
    float* hid   = xln   + nBLD;
    float* xz    = hid   + nBLD;                  // (B,L,2DI)
    float* xc    = xz    + 2 * nBLDI;
    float* proj  = xc    + nBLDI;                 // (B,L,80)
    float* delta = proj  + (size_t)kM * kPC;
    float* yb    = delta + nBLDI;

    // residual = 0 (fresh every call -> deterministic)
    zero_kernel<<<(unsigned)((nBLD + 255) / 256), 256, 0, stream>>>(resid, nBLD);

    const int nElem = (int)nBLDI;                 // B*L*DI elementwise count
    const float* hid_src = x;

    for (int i = 0; i < kNL; ++i) {
        // residual += hidden ; xln = LN(residual)
        add_layernorm<<<kM, 256, 0, stream>>>(hid_src, resid,
                                              ln_w + (size_t)i * kD,
                                              ln_b + (size_t)i * kD, xln);

        // xz = xln @ W_in   (2048x768)(768x3072)
        gemm32x32<kD, kD, 2 * kDI, 2 * kDI>
            <<<dim3(kM / 64, (2 * kDI) / 128), 256, 0, stream>>>(
                xln, W_in + (size_t)i * kD * 2 * kDI, xz);

        // xc = silu(conv(xz[...,:DI]) + conv_b)
        conv_silu<<<nElem / 256, 256, 0, stream>>>(xz,
            conv_w + (size_t)i * kDI * kK, conv_b + (size_t)i * kDI, xc);

        // proj = xc @ W_x   (2048x1536)(1536x80)
        gemm32x16<kPC, kDI, kDI, kPC, kPC>
            <<<dim3(kM / 64, (kPC + 63) / 64), 256, 0, stream>>>(
                xc, W_x + (size_t)i * kDI * kPC, proj);

        // delta_raw = dt @ W_dt   (dt = proj[:, :48], lda = 80)
        gemm32x32<kR, kPC, kDI, kDI>
            <<<dim3(kM / 64, kDI / 128), 256, 0, stream>>>(
                proj, W_dt + (size_t)i * kR * kDI, delta);

        // delta = softplus(delta_raw + b_dt)
        softplus_bias<<<nElem / 256, 256, 0, stream>>>(delta,
            b_dt + (size_t)i * kDI);

        // selective scan -> yb (includes +xc*D_skip)
        selective_scan<<<dim3(kDI / 256, kB), 256, 0, stream>>>(
            delta, xc, proj,
            A_log + (size_t)i * kDI * kN, D_skip + (size_t)i * kDI, yb);

        // yb *= silu(z)
        gate_silu<<<nElem / 256, 256, 0, stream>>>(yb, xz);

        // hid = yb @ W_out   (2048x1536)(1536x768)
        gemm32x32<kDI, kDI, kD, kD>
            <<<dim3(kM / 64, kD / 128), 256, 0, stream>>>(
                yb, W_out + (size_t)i * kDI * kD, hid);

        hid_src = hid;
    }

    // final: residual += hidden ; out = LN(residual, lnf)
    add_layernorm<<<kM, 256, 0, stream>>>(hid_src, resid, lnf_w, lnf_b, out);
}